// FlowReversal_6485400617539
// MI455X (gfx1250) — compile-verified
//
#include <hip/hip_runtime.h>
#include <stdint.h>

namespace {
constexpr int       kN   = 8;
constexpr int       kC   = 3;
constexpr int       kH   = 512;
constexpr int       kW   = 1024;
constexpr int       kHW  = kH * kW;                 // 524,288
constexpr long long kCHW = (long long)kC * kHW;     // 1,572,864
constexpr int       kPix = kN * kHW;                // 4,194,304
constexpr long long kOut = 2LL * kN * kCHW;         // 25,165,824 floats
}

// Non-returning device-scope FP32 atomic add straight to the L2 atomic units.
// TH defaults to 0 -> non-returning RT atomic, tracked with STOREcnt only.
__device__ __forceinline__ void gatomic_add_f32(float* p, float v) {
  asm volatile("global_atomic_add_f32 %0, %1, off scope:SCOPE_DEV"
               :: "v"(p), "v"(v)
               : "memory");
}

// Zero-init the (poisoned) output with 128-bit stores.
__global__ void fr_zero(float4* __restrict__ p, int n4) {
  int i = blockIdx.x * blockDim.x + threadIdx.x;
  if (i < n4) p[i] = make_float4(0.f, 0.f, 0.f, 0.f);
}

// One thread per source pixel: 5 coalesced b32 loads, 4 v_exp_f32,
// up to 16 global_atomic_add_f32 (4 corners x (3 channels + weight)).
__global__ void fr_splat(const float* __restrict__ img,
                         const float* __restrict__ flo,
                         float* __restrict__ imgw,   // d_out, NCHW
                         float* __restrict__ oacc) { // d_out + N*C*H*W (weight plane = ch0)
  int gid = blockIdx.x * blockDim.x + threadIdx.x;
  if (gid >= kPix) return;
  int n = gid / kHW;
  int p = gid - n * kHW;
  int h = p >> 10;        // p / W  (W == 1024)
  int w = p & (kW - 1);   // p % W

  const float* fbase = flo + (long long)n * 2 * kHW + p;
  float y = fbase[0];       // flo[n,0,h,w]  -> W axis
  float x = fbase[kHW];     // flo[n,1,h,w]  -> H axis

  const float* ibase = img + (long long)n * kCHW + p;
  float c0 = ibase[0];
  float c1 = ibase[kHW];
  float c2 = ibase[2 * kHW];

  float fx1 = floorf(x), fy1 = floorf(y);
  float ax1 = x - fx1,   ay1 = y - fy1;     // in [0,1)
  float ax2 = ax1 - 1.f, ay2 = ay1 - 1.f;
  float dx1 = ax1 * ax1, dx2 = ax2 * ax2;
  float dy1 = ay1 * ay1, dy2 = ay2 * ay2;

  const float kL2E = 1.4426950408889634f;   // log2(e): exp(-d) = exp2(-d*log2e)
  float w11 = __builtin_amdgcn_exp2f(-(dx1 + dy1) * kL2E);
  float w12 = __builtin_amdgcn_exp2f(-(dx1 + dy2) * kL2E);
  float w21 = __builtin_amdgcn_exp2f(-(dx2 + dy1) * kL2E);
  float w22 = __builtin_amdgcn_exp2f(-(dx2 + dy2) * kL2E);

  int ix1 = (int)fx1 + h;   // H-axis target
  int iy1 = (int)fy1 + w;   // W-axis target

  float* ibw = imgw + (long long)n * kCHW;
  float* obw = oacc + (long long)n * kCHW;  // channel-0 plane of `o`

  float wts[4] = {w11, w12, w21, w22};
#pragma unroll
  for (int k = 0; k < 4; ++k) {
    int ix = ix1 + (k >> 1);   // k=0,1 -> x1 ; k=2,3 -> x2
    int iy = iy1 + (k & 1);    // k=0,2 -> y1 ; k=1,3 -> y2
    float wt = wts[k];
    if (ix >= 0 && ix < kH && iy >= 0 && iy < kW) {
      long long t = (long long)ix * kW + iy;
      gatomic_add_f32(ibw + t,            c0 * wt);
      gatomic_add_f32(ibw + t + kHW,      c1 * wt);
      gatomic_add_f32(ibw + t + 2 * kHW,  c2 * wt);
      gatomic_add_f32(obw + t,            wt);
    }
  }
}

// Broadcast the accumulated weight plane (channel 0) to channels 1 and 2.
__global__ void fr_bcast(float* __restrict__ o) {
  int gid = blockIdx.x * blockDim.x + threadIdx.x;
  if (gid >= kPix) return;
  int n = gid / kHW;
  int p = gid - n * kHW;
  float* base = o + (long long)n * kCHW + p;
  float v = base[0];
  base[kHW]     = v;
  base[2 * kHW] = v;
}

extern "C" void kernel_launch(void* const* d_in, const int* in_sizes, int n_in,
                              void* d_out, int out_size, void* d_ws, size_t ws_size,
                              hipStream_t stream) {
  (void)in_sizes; (void)n_in; (void)out_size; (void)d_ws; (void)ws_size;

  const float* img = (const float*)d_in[0];   // (8,3,512,1024) f32
  const float* flo = (const float*)d_in[1];   // (8,2,512,1024) f32
  float* out  = (float*)d_out;
  float* imgw = out;                          // first output, NCHW
  float* o    = out + (long long)kN * kCHW;   // second output, NCHW

  constexpr int kBlk = 256;                   // 8 wave32 per block

  int n4 = (int)(kOut / 4);                   // out_size divisible by 4
  fr_zero<<<(n4 + kBlk - 1) / kBlk, kBlk, 0, stream>>>((float4*)out, n4);
  fr_splat<<<(kPix + kBlk - 1) / kBlk, kBlk, 0, stream>>>(img, flo, imgw, o);
  fr_bcast<<<(kPix + kBlk - 1) / kBlk, kBlk, 0, stream>>>(o);
}